// ChamferDistance_8701603742377
// MI455X (gfx1250) — compile-verified
//
#include <hip/hip_runtime.h>

typedef __attribute__((ext_vector_type(2))) float v2f;
typedef __attribute__((ext_vector_type(8))) float v8f;

#define FLT_BIG 3.0e38f

// Pre-pack a point set into WMMA-B-fragment order:
//   P0[j] = { x_j, z_j }          (K components 0 and 2: lanes 0..15)
//   P1[j] = { y_j, |p_j|^2 }      (K components 1 and 3: lanes 16..31)
// so the inner loop's B fragment is a single global_load_b64 per lane.
__global__ __launch_bounds__(256) void pack_kernel(
    const float* __restrict__ P, int n,
    v2f* __restrict__ out0, v2f* __restrict__ out1) {
  const int i = blockIdx.x * blockDim.x + threadIdx.x;
  if (i < n) {
    const float x = P[3 * i + 0];
    const float y = P[3 * i + 1];
    const float z = P[3 * i + 2];
    v2f e0, e1;
    e0.x = x;
    e0.y = z;
    e1.x = y;
    e1.y = x * x + y * y + z * z;
    out0[i] = e0;
    out1[i] = e1;
  }
}

// For each of 16 A-points per wave, compute min over all B points of squared
// Euclidean distance. One V_WMMA_F32_16X16X4_F32 per 16x16 tile:
//   A row m = {-2ax, -2ay, -2az, 1}     (loop-invariant)
//   B col n = { bx,   by,   bz, |b|^2}  (pre-packed: loads ARE the fragment)
//   C       = 0                          (inline SRC2=0)
//   D[m][n] = |b_n|^2 - 2 a_m.b_n  =: t
// min_n (|a|^2 + t_n) = |a|^2 + min_n t_n, and max(.,0) is monotone, so the
// row-constant |a|^2 and the zero clamp are applied once after the sweep.
__global__ __launch_bounds__(256) void nn_min16_kernel(
    const float* __restrict__ A, int nA,
    const v2f* __restrict__ B0, const v2f* __restrict__ B1, int nB,
    float* __restrict__ outMin) {
  const int lane = threadIdx.x & 31;
  const int wavesPerBlock = blockDim.x >> 5;
  const int waveId = blockIdx.x * wavesPerBlock + (threadIdx.x >> 5);
  const int mBase = waveId * 16;
  if (mBase >= nA) return;  // wave-uniform exit: EXEC stays all-ones for WMMA

  const int h = lane >> 4;    // wave half: selects which K components we hold
  const int idx = lane & 15;  // tile row (A) / tile col (B) this lane services

  // ---- Loop-invariant A fragment ----
  int ai = mBase + idx;
  if (ai > nA - 1) ai = nA - 1;  // duplicate clamp: min-neutral
  const float ax = A[3 * ai + 0];
  const float ay = A[3 * ai + 1];
  const float az = A[3 * ai + 2];
  const float sqA = ax * ax + ay * ay + az * az;

  // A 16x4 layout: lane holds M=idx, K components {2h, 2h+1} of
  // {-2ax, -2ay, -2az, 1}
  v2f afrag;
  afrag.x = h ? (-2.0f * az) : (-2.0f * ax);
  afrag.y = h ? 1.0f : (-2.0f * ay);

  // Running min of t = |b|^2 - 2 a.b for rows M = 8h + r
  float rm[8];
#pragma unroll
  for (int r = 0; r < 8; ++r) rm[r] = FLT_BIG;

  // ---- Sweep B in 16-wide tiles: fragment comes straight from the load ----
  const v2f* bp = (h ? B1 : B0) + idx;  // per-lane pointer, +16 entries/tile
  const int nFull = nB & ~15;
#pragma unroll 4
  for (int nBase = 0; nBase < nFull; nBase += 16, bp += 16) {
    const v2f bfrag = *bp;  // one global_load_b64: the whole B fragment

    v8f c = {};  // inline 0 accumulator
    v8f acc = __builtin_amdgcn_wmma_f32_16x16x4_f32(
        false, afrag, false, bfrag, (short)0, c, false, false);

#pragma unroll
    for (int r = 0; r < 8; ++r) rm[r] = fminf(rm[r], acc[r]);
  }

  // ---- Tail tile (only when nB % 16 != 0; wave-uniform branch) ----
  if (nFull < nB) {
    int bi = nFull + idx;
    if (bi > nB - 1) bi = nB - 1;  // duplicate clamp: min-neutral
    const v2f bfrag = (h ? B1 : B0)[bi];
    v8f c = {};
    v8f acc = __builtin_amdgcn_wmma_f32_16x16x4_f32(
        false, afrag, false, bfrag, (short)0, c, false, false);
#pragma unroll
    for (int r = 0; r < 8; ++r) rm[r] = fminf(rm[r], acc[r]);
  }

  // ---- Fold the 16 lanes of each half: butterfly min over columns ----
#pragma unroll
  for (int r = 0; r < 8; ++r) {
#pragma unroll
    for (int m = 8; m >= 1; m >>= 1)
      rm[r] = fminf(rm[r], __shfl_xor(rm[r], m, 32));
  }

  // |a|^2 of the row this lane will write (row = mBase + 8h + idx, idx<8):
  // held by lane 8h+idx (lanes 0..15 hold points 0..15 of the tile)
  const float sqA_row = __shfl(sqA, 8 * h + idx, 32);

  // lanes idx<8 of half h own rows M = 8h + idx  (exclusive: direct store)
  if (idx < 8) {
    float v = rm[0];
#pragma unroll
    for (int r = 1; r < 8; ++r) v = (idx == r) ? rm[r] : v;
    v = fmaxf(v + sqA_row, 0.0f);  // add row norm, clamp once
    const int row = mBase + 8 * h + idx;
    if (row < nA) outMin[row] = v;
  }
}

// Single-workgroup reduction: out = mean(sqrt(minA)) + mean(sqrt(minB))
__global__ __launch_bounds__(1024) void chamfer_reduce_kernel(
    const float* __restrict__ minA, int nA,
    const float* __restrict__ minB, int nB,
    float* __restrict__ out) {
  __shared__ float sA[1024];
  __shared__ float sB[1024];
  float accA = 0.0f, accB = 0.0f;
  for (int i = threadIdx.x; i < nA; i += blockDim.x) accA += sqrtf(minA[i]);
  for (int i = threadIdx.x; i < nB; i += blockDim.x) accB += sqrtf(minB[i]);
  sA[threadIdx.x] = accA;
  sB[threadIdx.x] = accB;
  __syncthreads();
  for (int s = blockDim.x >> 1; s > 0; s >>= 1) {
    if ((int)threadIdx.x < s) {
      sA[threadIdx.x] += sA[threadIdx.x + s];
      sB[threadIdx.x] += sB[threadIdx.x + s];
    }
    __syncthreads();
  }
  if (threadIdx.x == 0) out[0] = sA[0] / (float)nA + sB[0] / (float)nB;
}

extern "C" void kernel_launch(void* const* d_in, const int* in_sizes, int n_in,
                              void* d_out, int out_size, void* d_ws, size_t ws_size,
                              hipStream_t stream) {
  const float* pc1 = (const float*)d_in[0];
  const float* pc2 = (const float*)d_in[1];
  const int n1 = in_sizes[0] / 3;  // 16384
  const int n2 = in_sizes[1] / 3;  // 16384

  // Scratch layout (all 8-byte aligned):
  //   min1[n1], min2[n2]  : per-point min d^2
  //   Q10[n1], Q11[n1]    : pc1 packed as WMMA-B fragments (v2f each)
  //   Q20[n2], Q21[n2]    : pc2 packed as WMMA-B fragments
  float* min1 = (float*)d_ws;
  float* min2 = min1 + n1;
  v2f* q10 = (v2f*)(min2 + n2);
  v2f* q11 = q10 + n1;
  v2f* q20 = q11 + n1;
  v2f* q21 = q20 + n2;
  float* out = (float*)d_out;

  const int threads = 256;  // 8 waves per block
  const int wavesPerBlock = threads / 32;

  // Pack both sets into fragment order (B-side staging, read 1024x each)
  pack_kernel<<<(n1 + threads - 1) / threads, threads, 0, stream>>>(pc1, n1, q10, q11);
  pack_kernel<<<(n2 + threads - 1) / threads, threads, 0, stream>>>(pc2, n2, q20, q21);

  const int waves1 = (n1 + 15) / 16;
  const int waves2 = (n2 + 15) / 16;
  const int blocks1 = (waves1 + wavesPerBlock - 1) / wavesPerBlock;
  const int blocks2 = (waves2 + wavesPerBlock - 1) / wavesPerBlock;

  // min over pc2 for each pc1 point, and vice versa (distance is symmetric)
  nn_min16_kernel<<<blocks1, threads, 0, stream>>>(pc1, n1, q20, q21, n2, min1);
  nn_min16_kernel<<<blocks2, threads, 0, stream>>>(pc2, n2, q10, q11, n1, min2);
  chamfer_reduce_kernel<<<1, 1024, 0, stream>>>(min1, n1, min2, n2, out);
}